// GCN_4741643894901
// MI455X (gfx1250) — compile-verified
//
#include <hip/hip_runtime.h>
#include <hip/hip_bf16.h>

typedef float v2f __attribute__((ext_vector_type(2)));
typedef float v8f __attribute__((ext_vector_type(8)));

#define ASYNC_A_STAGE 1   // use global_load_async_to_lds_b128 for A-tile staging

// ---------------------------------------------------------------------------
// Utility kernels
// ---------------------------------------------------------------------------
__global__ void k_zero_f32(float* __restrict__ p, int n) {
    int i = blockIdx.x * blockDim.x + threadIdx.x;
    if (i < n) p[i] = 0.0f;
}

__global__ void k_zero_i32(int* __restrict__ p, int n) {
    int i = blockIdx.x * blockDim.x + threadIdx.x;
    if (i < n) p[i] = 0;
}

// degi[dst[e]] += 1 (int histogram; self-loop handled analytically later)
__global__ void k_deg_count(const int* __restrict__ dst, int* __restrict__ degi, int E) {
    int e = blockIdx.x * blockDim.x + threadIdx.x;
    if (e < E) atomicAdd(&degi[dst[e]], 1);
}

// dinv[i] = rsqrt(degi[i] + 1)   (+1 = self-loop; always > 0)
__global__ void k_dinv(const int* __restrict__ degi, float* __restrict__ dinv, int n) {
    int i = blockIdx.x * blockDim.x + threadIdx.x;
    if (i < n) dinv[i] = __frsqrt_rn((float)degi[i] + 1.0f);
}

// single-block exclusive scan of degi -> offs[0..n], offs[n] = E
__global__ __launch_bounds__(1024) void k_scan_offsets(
    const int* __restrict__ degi, int* __restrict__ offs, int n) {
    __shared__ int buf[1024];
    __shared__ int base;
    const int t = threadIdx.x;
    if (t == 0) base = 0;
    __syncthreads();
    for (int start = 0; start < n; start += 1024) {
        int i = start + t;
        int v = (i < n) ? degi[i] : 0;
        buf[t] = v;
        __syncthreads();
        for (int off = 1; off < 1024; off <<= 1) {      // Hillis-Steele inclusive
            int add = (t >= off) ? buf[t - off] : 0;
            __syncthreads();
            buf[t] += add;
            __syncthreads();
        }
        if (i < n) offs[i] = base + buf[t] - v;         // exclusive
        __syncthreads();                                 // all base reads done
        if (t == 0) base += buf[1023];
        __syncthreads();
    }
    if (t == 0) offs[n] = base;
}

__global__ void k_copy_i32(const int* __restrict__ a, int* __restrict__ b, int n) {
    int i = blockIdx.x * blockDim.x + threadIdx.x;
    if (i < n) b[i] = a[i];
}

// CSR fill: slot = cursor[dst[e]]++ ; csr_src[slot] = src[e]
__global__ void k_csr_fill(const int* __restrict__ src, const int* __restrict__ dst,
                           int* __restrict__ cursor, int* __restrict__ csr_src, int E) {
    int e = blockIdx.x * blockDim.x + threadIdx.x;
    if (e < E) {
        int pos = atomicAdd(&cursor[dst[e]], 1);
        csr_src[pos] = src[e];
    }
}

// ---------------------------------------------------------------------------
// Fused GCN aggregation (CSR gather, no float atomics):
//   h[i] = relu( bias + dinv[i]^2 * t[i] + sum_{s in N(i)} dinv[s]*dinv[i]*t[s] )
// 4 nodes per 256-thread block; 64 lanes (2 waves) per node; float4 per lane.
// t (50000x256 = 51 MB) is L2-resident on MI455X (192 MB L2) -> runs at L2
// load bandwidth instead of L2-atomic throughput.
// ---------------------------------------------------------------------------
__global__ __launch_bounds__(256) void k_gather_conv(
    const float* __restrict__ t, const float* __restrict__ dinv,
    const int* __restrict__ offs, const int* __restrict__ csr_src,
    const float* __restrict__ bias, float* __restrict__ h,
    int nNodes, int C) {
    const int tid = threadIdx.x;
    const int node = blockIdx.x * 4 + (tid >> 6);
    if (node >= nNodes) return;
    const int c4 = (tid & 63) << 2;                     // C == 256
    const float dn = dinv[node];

    const float4 tv = *(const float4*)(t + (size_t)node * C + c4);
    float4 sum;
    sum.x = dn * dn * tv.x;
    sum.y = dn * dn * tv.y;
    sum.z = dn * dn * tv.z;
    sum.w = dn * dn * tv.w;

    const int j1 = offs[node + 1];
    for (int j = offs[node]; j < j1; ++j) {
        int s = csr_src[j];
        float coef = dinv[s] * dn;
        const float4 v = *(const float4*)(t + (size_t)s * C + c4);
        sum.x = fmaf(coef, v.x, sum.x);
        sum.y = fmaf(coef, v.y, sum.y);
        sum.z = fmaf(coef, v.z, sum.z);
        sum.w = fmaf(coef, v.w, sum.w);
    }

    const float4 bv = *(const float4*)(bias + c4);
    float4 r;
    r.x = fmaxf(sum.x + bv.x, 0.0f);
    r.y = fmaxf(sum.y + bv.y, 0.0f);
    r.z = fmaxf(sum.z + bv.z, 0.0f);
    r.w = fmaxf(sum.w + bv.w, 0.0f);
    *(float4*)(h + (size_t)node * C + c4) = r;
}

// global max pool: values are post-ReLU (>=0), so uint-bit-pattern atomicMax
// on a zero-initialized buffer is exact and order-independent (deterministic).
__global__ void k_pool_max(const float* __restrict__ h, const int* __restrict__ batch,
                           float* __restrict__ g, int nNodes, int C) {
    int idx = blockIdx.x * blockDim.x + threadIdx.x;
    if (idx < nNodes * C) {
        int node = idx / C;
        int c = idx - node * C;
        int b = batch[node];
        atomicMax((unsigned int*)&g[(size_t)b * C + c], __float_as_uint(h[idx]));
    }
}

// ---------------------------------------------------------------------------
// fp32 WMMA GEMM (compile-time N, K):  C[M x N] = A[M x K] @ B[K x N]
// Row-major; M%16==0, N%128==0, K%4==0.  One wave -> one 16x16 tile; 8 waves
// per block (16 rows x 128 cols).  A tile staged in LDS via async copy
// (GLOBAL_LOAD_ASYNC_TO_LDS_B128, ASYNCcnt), +4-float row pad kills bank
// conflicts.  Compile-time N/K folds all k*N terms into immediate offsets.
// ---------------------------------------------------------------------------
template <int N, int K>
__global__ __launch_bounds__(256) void k_gemm_wmma_f32(
    const float* __restrict__ A, const float* __restrict__ B,
    float* __restrict__ C) {
    constexpr int KP = K + 4;
    __shared__ float As[16 * KP];
    const int r0 = blockIdx.x * 16;
    const int tid = threadIdx.x;

#if ASYNC_A_STAGE
    // 16*K floats in 16B chunks: 4*K chunks, 256 threads -> K/64 per thread.
    // Both global and LDS addresses are 16B aligned (K%4==0, KP*4 % 16 == 0).
#pragma unroll
    for (int it = 0; it < (4 * K) / 256; ++it) {
        int chunk4 = (it * 256 + tid) * 4;              // float index
        int row = chunk4 / K;
        int col = chunk4 % K;
        const float* gp = A + (size_t)(r0 + row) * K + col;
        unsigned int laddr =
            (unsigned int)(uintptr_t)(&As[row * KP + col]);  // low 32b = LDS offset
        asm volatile("global_load_async_to_lds_b128 %0, %1, off"
                     :: "v"(laddr), "v"((unsigned long long)(uintptr_t)gp)
                     : "memory");
    }
    asm volatile("s_wait_asynccnt 0x0" ::: "memory");
#else
    for (int idx = tid; idx < 16 * K; idx += 256) {     // coalesced A-tile load
        int row = idx / K;
        int col = idx - row * K;
        As[row * KP + col] = A[(size_t)(r0 + row) * K + col];
    }
#endif
    __syncthreads();

    const int wave = tid >> 5;
    const int lane = tid & 31;
    const int half = lane >> 4;                         // 0: K,K+1  1: K+2,K+3
    const int l    = lane & 15;
    const int c0   = (blockIdx.y * 8 + wave) * 16;

    // lanes 0-15: rows k,k+1 ; lanes 16-31: rows k+2,k+3 (fp32 16x4 A layout)
    const float* Bp = B + (size_t)(2 * half) * N + c0 + l;
    const float* Ap = &As[l * KP + 2 * half];

    v8f acc = {};
#pragma unroll
    for (int k = 0; k < K; k += 4) {
        v2f a = *(const v2f*)(Ap + k);
        v2f b;
        b.x = Bp[(size_t)k * N];                        // immediate offsets
        b.y = Bp[(size_t)(k + 1) * N];
        acc = __builtin_amdgcn_wmma_f32_16x16x4_f32(
            false, a, false, b, (short)0, acc, false, false);
    }

    float* Cp = C + (size_t)(r0 + 8 * half) * N + c0 + l;
#pragma unroll
    for (int j = 0; j < 8; ++j)
        Cp[(size_t)j * N] = acc[j];
}

// ---------------------------------------------------------------------------
// Fused FC head: per graph  g(256) -> relu(fc1: 128) -> fc2: 8 -> log_softmax
// ---------------------------------------------------------------------------
__global__ __launch_bounds__(128) void k_fc_head(
    const float* __restrict__ g,
    const float* __restrict__ fcW1, const float* __restrict__ fcb1,
    const float* __restrict__ fcW2, const float* __restrict__ fcb2,
    float* __restrict__ out, int HIDc, int H2, int NC) {
    __shared__ float sg[256];
    __shared__ float sh[128];
    __shared__ float sl[8];
    const int gid = blockIdx.x;
    const int t = threadIdx.x;

    for (int k = t; k < HIDc; k += H2) sg[k] = g[(size_t)gid * HIDc + k];
    __syncthreads();

    float acc = fcb1[t];
    for (int k = 0; k < HIDc; ++k) acc = fmaf(sg[k], fcW1[(size_t)k * H2 + t], acc);
    sh[t] = fmaxf(acc, 0.0f);
    __syncthreads();

    if (t < NC) {
        float a = fcb2[t];
        for (int k = 0; k < H2; ++k) a = fmaf(sh[k], fcW2[(size_t)k * NC + t], a);
        sl[t] = a;
    }
    __syncthreads();

    if (t == 0) {
        float m = sl[0];
        for (int c = 1; c < NC; ++c) m = fmaxf(m, sl[c]);
        float s = 0.0f;
        for (int c = 0; c < NC; ++c) s += __expf(sl[c] - m);
        float lse = __logf(s);
        for (int c = 0; c < NC; ++c) out[(size_t)gid * NC + c] = sl[c] - m - lse;
    }
}

// ---------------------------------------------------------------------------
extern "C" void kernel_launch(void* const* d_in, const int* in_sizes, int n_in,
                              void* d_out, int out_size, void* d_ws, size_t ws_size,
                              hipStream_t stream) {
    const float* x     = (const float*)d_in[0];
    const int*   eidx  = (const int*)  d_in[1];
    const int*   batch = (const int*)  d_in[2];
    const float* W1    = (const float*)d_in[3];
    const float* b1    = (const float*)d_in[4];
    const float* W2    = (const float*)d_in[5];
    const float* b2    = (const float*)d_in[6];
    const float* fcW1  = (const float*)d_in[7];
    const float* fcb1  = (const float*)d_in[8];
    const float* fcW2  = (const float*)d_in[9];
    const float* fcb2  = (const float*)d_in[10];
    float* out = (float*)d_out;

    const int nNodes = in_sizes[2];            // 50000
    const int HID    = in_sizes[4];            // 256
    const int E      = in_sizes[1] / 2;        // 600000
    const int H2     = in_sizes[8];            // 128
    const int NC     = in_sizes[10];           // 8
    const int nGr    = out_size / NC;          // 512

    const int* src = eidx;
    const int* dst = eidx + E;

    // ---- workspace layout ----
    float* wsf  = (float*)d_ws;
    size_t o = 0;
    float* dinv = wsf + o; o += ((size_t)nNodes + 255) & ~(size_t)255;
    float* gpl  = wsf + o; o += (size_t)nGr * HID;
    float* bufT = wsf + o; o += (size_t)nNodes * HID;   // GEMM output t
    float* bufA = wsf + o; o += (size_t)nNodes * HID;   // aggregated h
    int* wsi = (int*)(wsf + o);
    size_t oi = 0;
    int* degi    = wsi + oi; oi += (size_t)nNodes;
    int* offs    = wsi + oi; oi += (size_t)nNodes + 1;
    int* cursor  = wsi + oi; oi += (size_t)nNodes;
    int* csr_src = wsi + oi;

    const int TB = 256;
    const int gridN  = (nNodes + TB - 1) / TB;
    const int gridE  = (E + TB - 1) / TB;
    const int gridNC = (nNodes * HID + TB - 1) / TB;
    const int gridGa = (nNodes + 3) / 4;               // 4 nodes / block

    // 1) degree -> dinv ; CSR build (reused by both layers)
    k_zero_i32<<<gridN, TB, 0, stream>>>(degi, nNodes);
    k_deg_count<<<gridE, TB, 0, stream>>>(dst, degi, E);
    k_dinv<<<gridN, TB, 0, stream>>>(degi, dinv, nNodes);
    k_scan_offsets<<<1, 1024, 0, stream>>>(degi, offs, nNodes);
    k_copy_i32<<<gridN, TB, 0, stream>>>(offs, cursor, nNodes);
    k_csr_fill<<<gridE, TB, 0, stream>>>(src, dst, cursor, csr_src, E);

    // 2) layer 1:  t = x @ W1 ;  h1 = relu(aggregate(t) + b1)
    dim3 g1((unsigned)(nNodes / 16), 2u);              // HID/128 = 2
    k_gemm_wmma_f32<256, 128><<<g1, TB, 0, stream>>>(x, W1, bufT);
    k_gather_conv<<<gridGa, TB, 0, stream>>>(bufT, dinv, offs, csr_src, b1, bufA, nNodes, HID);

    // 3) layer 2:  t = h1 @ W2 ;  h2 = relu(aggregate(t) + b2)
    k_gemm_wmma_f32<256, 256><<<g1, TB, 0, stream>>>(bufA, W2, bufT);
    k_gather_conv<<<gridGa, TB, 0, stream>>>(bufT, dinv, offs, csr_src, b2, bufA, nNodes, HID);

    // 4) global max pool
    k_zero_f32<<<((nGr * HID) + TB - 1) / TB, TB, 0, stream>>>(gpl, nGr * HID);
    k_pool_max<<<gridNC, TB, 0, stream>>>(bufA, batch, gpl, nNodes, HID);

    // 5) FC head + log_softmax
    k_fc_head<<<nGr, H2, 0, stream>>>(gpl, fcW1, fcb1, fcW2, fcb2, out, HID, H2, NC);
}